// KGEModel_50096498540662
// MI455X (gfx1250) — compile-verified
//
#include <hip/hip_runtime.h>

#define K_B        512
#define K_NEG      256
#define K_HID      100
#define K_HN       10
#define K_GAMMA    12.0f
#define K_THRED    0.5f

typedef float v2f __attribute__((ext_vector_type(2)));
typedef float v8f __attribute__((ext_vector_type(8)));

__device__ __forceinline__ void reflect4(float h[4], const float* v, float c) {
    float d = v[0]*h[0] + v[1]*h[1] + v[2]*h[2] + v[3]*h[3];
    float cd = c * d;
    h[0] -= cd*v[0]; h[1] -= cd*v[1]; h[2] -= cd*v[2]; h[3] -= cd*v[3];
}

#if __has_builtin(__builtin_amdgcn_permlane16)
__device__ __forceinline__ float permx(float x, unsigned lo, unsigned hi) {
    unsigned u = __float_as_uint(x);
    unsigned r = __builtin_amdgcn_permlane16(u, u, lo, hi, false, false);
    return __uint_as_float(r);
}
// sel[i] = i ^ m, packed as 16 nibbles into {hi,lo}
__device__ __forceinline__ float lxor1(float x) { return permx(x, 0x67452301u, 0xEFCDAB89u); }
__device__ __forceinline__ float lxor2(float x) { return permx(x, 0x54761032u, 0xDCFE98BAu); }
__device__ __forceinline__ float lxor4(float x) { return permx(x, 0x32107654u, 0xBA98FEDCu); }
__device__ __forceinline__ float lxor8(float x) { return permx(x, 0xFEDCBA98u, 0x76543210u); }
#else
__device__ __forceinline__ float lxor1(float x) { return __shfl_xor(x, 1, 32); }
__device__ __forceinline__ float lxor2(float x) { return __shfl_xor(x, 2, 32); }
__device__ __forceinline__ float lxor4(float x) { return __shfl_xor(x, 4, 32); }
__device__ __forceinline__ float lxor8(float x) { return __shfl_xor(x, 8, 32); }
#endif

// One block per batch element b. 256 threads = 8 waves (wave32).
// Phase 1: per-hid precompute of head' (8 reflections) and the composed tail
//          reflection matrix M = R1*R0 (4x4) into LDS.
// Phase 2: 4 hids per chained-WMMA group; A loads software-pipelined one group
//          ahead. Chain element j: A = 16 tails of hid 4g+j (K=4), B nonzero
//          only in column quad j (= M_{4g+j}^T). D dense: col n = component
//          (n&3) of M*t for hid 4g+(n>>2).
__global__ __launch_bounds__(256) void house_score_kernel(
    const float* __restrict__ ent,  const float* __restrict__ rel,
    const float* __restrict__ kdh,  const float* __restrict__ kdt,
    const float* __restrict__ ksh,  const float* __restrict__ kst,
    const int*   __restrict__ head_idx, const int* __restrict__ rel_idx,
    const int*   __restrict__ tail_idx, float* __restrict__ out)
{
    __shared__ float sH[K_HID][4];    // head' per hid
    __shared__ float sM[K_HID][16];   // M[d][e] row-major per hid

    const int b   = blockIdx.x;
    const int tid = threadIdx.x;
    const int rid = rel_idx[b];
    const int he  = head_idx[b];

    // ---- Phase 1: precompute per-hid quantities ----
    for (int hid = tid; hid < K_HID; hid += blockDim.x) {
        float v[K_HN][4];
        const float* rp = rel + ((size_t)rid * K_HID + hid) * (K_HN * 4);
        #pragma unroll
        for (int j = 0; j < K_HN; ++j) {
            float x0 = rp[j*4+0], x1 = rp[j*4+1], x2 = rp[j*4+2], x3 = rp[j*4+3];
            float nrm = sqrtf(x0*x0 + x1*x1 + x2*x2 + x3*x3);
            float inv = 1.0f / fmaxf(nrm, 1e-12f);
            v[j][0] = x0*inv; v[j][1] = x1*inv; v[j][2] = x2*inv; v[j][3] = x3*inv;
        }
        float kh0 = fminf(kdh[rid*2+0] * fabsf(ksh[((size_t)rid*K_HID+hid)*2+0]), K_THRED);
        float kh1 = fminf(kdh[rid*2+1] * fabsf(ksh[((size_t)rid*K_HID+hid)*2+1]), K_THRED);
        float kt0 = fminf(kdt[rid*2+0] * fabsf(kst[((size_t)rid*K_HID+hid)*2+0]), K_THRED);
        float kt1 = fminf(kdt[rid*2+1] * fabsf(kst[((size_t)rid*K_HID+hid)*2+1]), K_THRED);

        float h[4];
        const float* hp = ent + ((size_t)he * K_HID + hid) * 4;
        h[0]=hp[0]; h[1]=hp[1]; h[2]=hp[2]; h[3]=hp[3];
        reflect4(h, v[9], kh0);
        reflect4(h, v[8], kh1);
        #pragma unroll
        for (int j = 7; j >= 2; --j) reflect4(h, v[j], 2.0f);
        sH[hid][0]=h[0]; sH[hid][1]=h[1]; sH[hid][2]=h[2]; sH[hid][3]=h[3];

        // M = R1 * R0 by columns: col e = R1(R0(basis_e))
        #pragma unroll
        for (int e = 0; e < 4; ++e) {
            float x[4];
            #pragma unroll
            for (int d = 0; d < 4; ++d)
                x[d] = (d == e ? 1.0f : 0.0f) - kt0 * v[0][e] * v[0][d];
            float dot = v[1][0]*x[0] + v[1][1]*x[1] + v[1][2]*x[2] + v[1][3]*x[3];
            #pragma unroll
            for (int d = 0; d < 4; ++d)
                sM[hid][d*4 + e] = x[d] - kt1 * dot * v[1][d];
        }
    }
    __syncthreads();

    // ---- Phase 2: scoring ----
    const int wave = tid >> 5;
    const int lane = tid & 31;
    const int n    = lane & 15;   // A row (neg in tile) / B,D column
    const int hl   = lane >> 4;   // lane half: K pair {0,1} vs {2,3}
    const int jq   = n >> 2;      // which hid of the 4-group this column serves
    const int e    = n & 3;       // component within house_dim

    const int NG = K_HID / 4;     // 25 groups of 4 hids

    for (int tile = wave; tile < 16; tile += 8) {
        const int    neg = tile * 16 + n;
        const size_t te  = (size_t)tail_idx[(size_t)b * K_NEG + neg];
        const float* tp  = ent + te * (K_HID * 4) + hl * 2;

        float acc[8];
        #pragma unroll
        for (int r = 0; r < 8; ++r) acc[r] = 0.0f;

        // pipeline prologue: A fragments for group 0
        float2 a0 = *(const float2*)(tp + 0);
        float2 a1 = *(const float2*)(tp + 4);
        float2 a2 = *(const float2*)(tp + 8);
        float2 a3 = *(const float2*)(tp + 12);

        for (int g = 0; g < NG; ++g) {
            // prefetch next group's A fragments (uniform clamp, no OOB)
            const int gn = (g < NG - 1) ? g + 1 : NG - 1;
            const float* tn = tp + gn * 16;
            const float2 p0 = *(const float2*)(tn + 0);
            const float2 p1 = *(const float2*)(tn + 4);
            const float2 p2 = *(const float2*)(tn + 8);
            const float2 p3 = *(const float2*)(tn + 12);

            const int h0 = g * 4;
            v2f A0; A0.x=a0.x; A0.y=a0.y;
            v2f A1; A1.x=a1.x; A1.y=a1.y;
            v2f A2; A2.x=a2.x; A2.y=a2.y;
            v2f A3; A3.x=a3.x; A3.y=a3.y;

            // This lane's own B values: B[k][n] = M_{h0+jq}[e][k], k = 2*hl+{0,1}
            const float2 m2 = *(const float2*)(&sM[h0 + jq][e*4 + 2*hl]);
            const float  hv = sH[h0 + jq][e];

            v2f B0; B0.x = (jq==0)? m2.x : 0.0f; B0.y = (jq==0)? m2.y : 0.0f;
            v2f B1; B1.x = (jq==1)? m2.x : 0.0f; B1.y = (jq==1)? m2.y : 0.0f;
            v2f B2; B2.x = (jq==2)? m2.x : 0.0f; B2.y = (jq==2)? m2.y : 0.0f;
            v2f B3; B3.x = (jq==3)? m2.x : 0.0f; B3.y = (jq==3)? m2.y : 0.0f;

            v8f C = {0.f,0.f,0.f,0.f,0.f,0.f,0.f,0.f};
            v8f D;
            D = __builtin_amdgcn_wmma_f32_16x16x4_f32(false, A0, false, B0, (short)0, C, false, false);
            D = __builtin_amdgcn_wmma_f32_16x16x4_f32(false, A1, false, B1, (short)0, D, false, false);
            D = __builtin_amdgcn_wmma_f32_16x16x4_f32(false, A2, false, B2, (short)0, D, false, false);
            D = __builtin_amdgcn_wmma_f32_16x16x4_f32(false, A3, false, B3, (short)0, D, false, false);

            #pragma unroll
            for (int r = 0; r < 8; ++r) {
                float dd = D[r] - hv;          // diff comp e, hid h0+jq, row r(/r+8)
                float sq = dd * dd;
                sq += lxor1(sq);               // reduce over the 4 components
                sq += lxor2(sq);               // (stays within each column quad)
                acc[r] += __builtin_amdgcn_sqrtf(sq);
            }

            a0 = p0; a1 = p1; a2 = p2; a3 = p3;   // rotate pipeline
        }

        // combine the 4 hid-quads (lanes n, n^4, n^8 hold disjoint hid subsets)
        #pragma unroll
        for (int r = 0; r < 8; ++r) {
            acc[r] += lxor4(acc[r]);
            acc[r] += lxor8(acc[r]);
        }

        if (n == 0) {  // lane 0 -> rows r, lane 16 -> rows r+8
            #pragma unroll
            for (int r = 0; r < 8; ++r)
                out[(size_t)b * K_NEG + tile * 16 + r + 8 * hl] = K_GAMMA - acc[r];
        }
    }
}

extern "C" void kernel_launch(void* const* d_in, const int* in_sizes, int n_in,
                              void* d_out, int out_size, void* d_ws, size_t ws_size,
                              hipStream_t stream) {
    const float* ent = (const float*)d_in[0];
    const float* rel = (const float*)d_in[1];
    const float* kdh = (const float*)d_in[2];
    const float* kdt = (const float*)d_in[3];
    const float* ksh = (const float*)d_in[4];
    const float* kst = (const float*)d_in[5];
    const int* head_idx = (const int*)d_in[6];
    const int* rel_idx  = (const int*)d_in[7];
    const int* tail_idx = (const int*)d_in[8];
    float* out = (float*)d_out;

    hipLaunchKernelGGL(house_score_kernel, dim3(K_B), dim3(256), 0, stream,
                       ent, rel, kdh, kdt, ksh, kst, head_idx, rel_idx, tail_idx, out);
}